// LBPSemanticDependency_36653250904394
// MI455X (gfx1250) — compile-verified
//
#include <hip/hip_runtime.h>
#include <math.h>

// Problem constants (B=2, S=160, K=2 labels, MAX_ITER=3)
constexpr int   cS   = 160;
constexpr int   cS2  = 160 * 160;            // 25600 (one (x,y) slice)
constexpr long  cS3  = 160L * 160L * 160L;   // 4,096,000
constexpr long  KSTR = 2L * cS3;             // k-stride inside a type: 8,192,000 floats
constexpr long  TSTR = 2L * KSTR;            // type stride: 16,384,000 floats
constexpr long  BRAW_OFF = 3L * TSTR;        // 49,152,000
constexpr int   KB2  = 2 * cS2;              // 51,200 (k-stride in braw/bel)
constexpr long  BEL_OFF  = BRAW_OFF + 2L * KB2;
// total ws floats: 49,356,800  (~197.5 MB)

typedef float v2f __attribute__((ext_vector_type(2)));
typedef float v8f __attribute__((ext_vector_type(8)));

__device__ __forceinline__ float lae(float u, float v) {
    float hi = fmaxf(u, v), lo = fminf(u, v);
    return hi + log1pf(expf(lo - hi));   // logaddexp, safe: lo-hi <= 0
}

// bel[k,batch,a,c] = log_softmax_k(braw) ; first iter: braw==0 -> -ln2
__global__ void prep_bel_kernel(const float* __restrict__ braw,
                                float* __restrict__ bel, int first) {
    int i = blockIdx.x * blockDim.x + threadIdx.x;
    if (i >= KB2) return;
    if (first) {
        bel[i] = -0.69314718055994531f;
        bel[i + KB2] = -0.69314718055994531f;
        return;
    }
    float u = braw[i], v = braw[i + KB2];
    float l = lae(u, v);
    bel[i] = u - l;
    bel[i + KB2] = v - l;
}

// Odd-parity update (iterations 1 and 3): S_new(x,y)=m_new(y,x)
//   value at (b=y,c=x): f from bel[c=x], m_prev(x,y)=S_prev(x,y) (in place), p=s[batch,x,a,y]
__global__ void update_odd_kernel(const float* __restrict__ s_sib,
                                  const float* __restrict__ s_cop,
                                  const float* __restrict__ s_grd,
                                  float* __restrict__ wsf, int first) {
    int bx = blockIdx.x;
    int type = bx / 320, rem = bx % 320, batch = rem / 160, a = rem % 160;
    const float* sc = (type == 0) ? s_sib : (type == 1) ? s_cop : s_grd;
    float* M0 = wsf + (size_t)type * TSTR + (size_t)(batch * 160 + a) * cS2;
    float* M1 = M0 + KSTR;
    const float* bel0 = wsf + BEL_OFF + batch * cS2 + a * cS;
    const float* bel1 = bel0 + KB2;
    for (int i = threadIdx.x; i < cS2; i += blockDim.x) {
        int x = i / 160, y = i - x * 160;
        float p = sc[(size_t)(batch * 160 + x) * cS2 + a * cS + y];
        float mp0 = 0.f, mp1 = 0.f;
        if (!first) { mp0 = M0[i]; mp1 = M1[i]; }
        float q0 = bel0[x] - mp0, q1 = bel1[x] - mp1;
        float f0 = lae(q0, q1), f1 = lae(q0, q1 + p);
        float l = lae(f0, f1);
        M0[i] = f0 - l;
        M1[i] = f1 - l;
    }
}

// Even-parity update (iteration 2): S_new(x,y)=m_new(x,y)
//   value at (b=x,c=y): f from bel[c=y], m_prev=S_prev(x,y) (in place), p=s[batch,y,a,x]
//   -> stage the 160x160 score slice in LDS (pad stride 161: conflict-free transpose read)
__global__ void update_even_kernel(const float* __restrict__ s_sib,
                                   const float* __restrict__ s_cop,
                                   const float* __restrict__ s_grd,
                                   float* __restrict__ wsf) {
    __shared__ float sT[160 * 161];   // 103,040 bytes (< 320 KB/WGP)
    int bx = blockIdx.x;
    int type = bx / 320, rem = bx % 320, batch = rem / 160, a = rem % 160;
    const float* sc = (type == 0) ? s_sib : (type == 1) ? s_cop : s_grd;
    float* M0 = wsf + (size_t)type * TSTR + (size_t)(batch * 160 + a) * cS2;
    float* M1 = M0 + KSTR;
    const float* bel0 = wsf + BEL_OFF + batch * cS2 + a * cS;
    const float* bel1 = bel0 + KB2;
    for (int i = threadIdx.x; i < cS2; i += blockDim.x) {
        int r = i / 160, c = i - r * 160;
        sT[r * 161 + c] = sc[(size_t)(batch * 160 + r) * cS2 + a * cS + c];
    }
    __syncthreads();
    for (int i = threadIdx.x; i < cS2; i += blockDim.x) {
        int x = i / 160, y = i - x * 160;
        float p = sT[y * 161 + x];               // s[batch, y, a, x]
        float mp0 = M0[i], mp1 = M1[i];
        float q0 = bel0[y] - mp0, q1 = bel1[y] - mp1;
        float f0 = lae(q0, q1), f1 = lae(q0, q1 + p);
        float l = lae(f0, f1);
        M0[i] = f0 - l;
        M1[i] = f1 - l;
    }
}

// braw[k,a,b,batch] = p_edge + maskO(b) * sum_types( sum_c M(b,c)*w1(c) - M(b,b)*w1(b) )
// Matvec on the matrix unit: V_WMMA_F32_16X16X4_F32 with A = data*weights, B = ones.
// TRANSPOSED=true: m(b,c)=S(c,b);  false: m(b,c)=S(b,c)
// Templated so each instantiation gets immediate-offset addressing and no per-load selects.
template <bool TRANSPOSED>
__global__ void reduce_kernel(const float* __restrict__ s_edge,
                              const unsigned char* __restrict__ mask,
                              float* __restrict__ wsf) {
    __shared__ float w1s[160];
    __shared__ float acc[320];
    int bx = blockIdx.x;
    int batch = bx / 160, a = bx % 160;
    int tid = threadIdx.x;
    if (tid < 160) {
        int c = tid;
        w1s[c] = (mask[(batch * 160 + c) * 160 + a] && (c != a)) ? 1.0f : 0.0f;
    }
    if (tid < 320) acc[tid] = 0.0f;
    __syncthreads();

    int wave = tid >> 5, lane = tid & 31;
    int ln = lane & 15;
    int kb = (lane < 16) ? 0 : 2;    // A-matrix 16x4 layout: VGPR0=K0/K2, VGPR1=K1/K3
    v2f Bv = {1.0f, 1.0f};           // B = all-ones: layout-permutation invariant
    for (int u = wave; u < 60; u += 8) {          // 3 types * 2 k * 10 row-tiles
        int type = u / 20, r = u % 20, k = r / 10, n0 = (r % 10) * 16;
        const float* Mt = wsf + (size_t)type * TSTR + (size_t)k * KSTR +
                          (size_t)(batch * 160 + a) * cS2;
        // Hoist the lane-dependent part of the address out of the K loop.
        const float* baseT = Mt + kb * 160 + n0 + ln;       // TRANSPOSED path
        const float* baseI = Mt + (n0 + ln) * 160 + kb;     // identity path
        v8f d = {0.f, 0.f, 0.f, 0.f, 0.f, 0.f, 0.f, 0.f};
        #pragma unroll
        for (int c0 = 0; c0 < 160; c0 += 4) {
            float w0 = w1s[c0 + kb], w1v = w1s[c0 + kb + 1];
            v2f A;
            if (TRANSPOSED) {        // A[m,kk] = S(c0+kk, n0+m) * w1(c0+kk)
                A.x = baseT[c0 * 160] * w0;
                A.y = baseT[c0 * 160 + 160] * w1v;
            } else {                 // A[m,kk] = S(n0+m, c0+kk) * w1(c0+kk)
                v2f mv = *(const v2f*)(baseI + c0);         // 8B-aligned: c0+kb even
                A.x = mv.x * w0;
                A.y = mv.y * w1v;
            }
            d = __builtin_amdgcn_wmma_f32_16x16x4_f32(false, A, false, Bv,
                                                      (short)0, d, false, false);
        }
        // D[m,n] constant over n; out[m]: m=0..7 -> d[m] lane 0, m=8..15 -> d[m-8] lane 16
        if (lane == 0 || lane == 16) {
            int mb = (lane == 0) ? 0 : 8;
            #pragma unroll
            for (int r8 = 0; r8 < 8; ++r8) {
                int m = n0 + mb + r8;
                float val = d[r8] - Mt[m * 160 + m] * w1s[m];   // exclude c == b
                atomicAdd(&acc[k * 160 + m], val);
            }
        }
    }
    __syncthreads();
    if (tid < 320) {
        int k = tid / 160, b = tid - k * 160;
        float mo = mask[(batch * 160 + b) * 160 + a] ? 1.0f : 0.0f;
        float pe = (k == 1) ? s_edge[(batch * 160 + b) * 160 + a] : 0.0f;
        wsf[BRAW_OFF + (size_t)k * KB2 + batch * cS2 + a * cS + b] =
            pe + acc[k * 160 + b] * mo;
    }
}

// out[batch, b, a, k] = exp(log_softmax_k braw[k, a, b, batch])
__global__ void final_kernel(const float* __restrict__ braw, float* __restrict__ out) {
    int i = blockIdx.x * blockDim.x + threadIdx.x;
    if (i >= KB2) return;                        // i = batch*S2 + b*S + a
    int batch = i / cS2, r2 = i - batch * cS2;
    int b = r2 / cS, a = r2 - b * cS;
    int ib = batch * cS2 + a * cS + b;
    float u = braw[ib], v = braw[ib + KB2];
    float l = lae(u, v);
    size_t o = (size_t)i * 2;
    out[o]     = expf(u - l);
    out[o + 1] = expf(v - l);
}

extern "C" void kernel_launch(void* const* d_in, const int* in_sizes, int n_in,
                              void* d_out, int out_size, void* d_ws, size_t ws_size,
                              hipStream_t stream) {
    const float* s_edge = (const float*)d_in[0];
    const float* s_sib  = (const float*)d_in[1];
    const float* s_cop  = (const float*)d_in[2];
    const float* s_grd  = (const float*)d_in[3];
    const unsigned char* mask = (const unsigned char*)d_in[4];
    float* wsf  = (float*)d_ws;
    float* out  = (float*)d_out;
    const float* braw = wsf + BRAW_OFF;
    float* bel  = wsf + BEL_OFF;

    for (int it = 0; it < 3; ++it) {
        prep_bel_kernel<<<(KB2 + 255) / 256, 256, 0, stream>>>(braw, bel, it == 0 ? 1 : 0);
        if (it == 1)
            update_even_kernel<<<960, 256, 0, stream>>>(s_sib, s_cop, s_grd, wsf);
        else
            update_odd_kernel<<<960, 256, 0, stream>>>(s_sib, s_cop, s_grd, wsf,
                                                       it == 0 ? 1 : 0);
        if (it != 1)
            reduce_kernel<true><<<320, 256, 0, stream>>>(s_edge, mask, wsf);
        else
            reduce_kernel<false><<<320, 256, 0, stream>>>(s_edge, mask, wsf);
    }
    final_kernel<<<(KB2 + 255) / 256, 256, 0, stream>>>(braw, out);
}